// DenseEquivariantFFT_16277926051926
// MI455X (gfx1250) — compile-verified
//
#include <hip/hip_runtime.h>
#include <hip/hip_bf16.h>

// ---------------- problem constants ----------------
static constexpr int BATCH = 1024;
static constexpr int CIN   = 32;
static constexpr int COUT  = 32;
static constexpr int NS    = 8;     // n_symm
static constexpr int GDIM  = 512;   // group size
static constexpr int KDIM  = 256;   // Cin * n_symm  (GEMM K)
static constexpr int NDIM  = 256;   // Cout * n_symm (GEMM N)
static constexpr int FREQ  = 64;    // 8x8 lattice frequencies
static constexpr int NCAN  = 34;    // canonical frequencies (real-input symmetry)

// Canonical frequency list: f = f1*8+f2 with f <= conj(f), conj = ((8-f1)&7, (8-f2)&7)
static constexpr int FCANON[NCAN] = {
    0, 1, 2, 3, 4,
    8, 9, 10, 11, 12, 13, 14, 15,
    16, 17, 18, 19, 20, 21, 22, 23,
    24, 25, 26, 27, 28, 29, 30, 31,
    32, 33, 34, 35, 36};

// 8-point twiddles: cos/sin(2*pi*j/8); constexpr => folded into the fully
// unrolled DFT loops at compile time.
static constexpr float C8[8] = {1.f, 0.70710678118654752f, 0.f, -0.70710678118654752f,
                                -1.f, -0.70710678118654752f, 0.f, 0.70710678118654752f};
static constexpr float S8[8] = {0.f, 0.70710678118654752f, 1.f, 0.70710678118654752f,
                                0.f, -0.70710678118654752f, -1.f, -0.70710678118654752f};

typedef __attribute__((ext_vector_type(2))) float v2f;
typedef __attribute__((ext_vector_type(8))) float v8f;
typedef int v4i __attribute__((vector_size(16)));   // matches builtin param type

#if __has_builtin(__builtin_amdgcn_global_load_async_to_lds_b128) && \
    __has_builtin(__builtin_amdgcn_s_wait_asynccnt)
#define USE_ASYNC_LDS 1
#else
#define USE_ASYNC_LDS 0
#endif

#if USE_ASYNC_LDS
#define GL_ASYNC_B128(gsrc, ldst)                                              \
  __builtin_amdgcn_global_load_async_to_lds_b128(                              \
      (__attribute__((address_space(1))) v4i*)(uintptr_t)(gsrc),               \
      (__attribute__((address_space(3))) v4i*)(uint32_t)(uintptr_t)(ldst),     \
      0, 0)
#endif

__device__ __forceinline__ int conj_freq(int f) {   // constant-folds in unrolled loops
  const int f1 = f >> 3, f2 = f & 7;
  return (((8 - f1) & 7) << 3) | ((8 - f2) & 7);
}

// ---------------------------------------------------------------------------
// Kernel 1: forward 2D DFT of x; store only canonical frequencies.
// Output planes xf_re/xf_im: [f][b][k], k = ci*8+s  (A-matrix, row-major).
// ---------------------------------------------------------------------------
__global__ __launch_bounds__(256) void k_fft_x(const float* __restrict__ x,
                                               float* __restrict__ xf_re,
                                               float* __restrict__ xf_im) {
  const int t  = blockIdx.x * 256 + threadIdx.x;  // 0 .. B*256-1
  const int b  = t >> 8;
  const int k  = t & 255;        // == ci*8 + s
  const int ci = k >> 3;
  const int s  = k & 7;
  const float* xp = x + ((size_t)b * CIN + ci) * GDIM + s;  // stride 8 over cells

  float tre[8][8], tim[8][8];
#pragma unroll
  for (int l1 = 0; l1 < 8; ++l1) {
    float row[8];
#pragma unroll
    for (int l2 = 0; l2 < 8; ++l2) row[l2] = xp[(size_t)(l1 * 8 + l2) * 8];
#pragma unroll
    for (int f2 = 0; f2 < 8; ++f2) {
      float ar = 0.f, ai = 0.f;
#pragma unroll
      for (int l2 = 0; l2 < 8; ++l2) {
        const int j = (l2 * f2) & 7;
        ar += row[l2] * C8[j];
        ai -= row[l2] * S8[j];
      }
      tre[l1][f2] = ar; tim[l1][f2] = ai;
    }
  }
#pragma unroll
  for (int f1 = 0; f1 < 8; ++f1) {
#pragma unroll
    for (int f2 = 0; f2 < 8; ++f2) {
      const int f = f1 * 8 + f2;
      if (f <= conj_freq(f)) {  // canonical only (folds; non-canonical DCE'd)
        float ar = 0.f, ai = 0.f;
#pragma unroll
        for (int l1 = 0; l1 < 8; ++l1) {
          const int j = (l1 * f1) & 7;
          ar += tre[l1][f2] * C8[j] + tim[l1][f2] * S8[j];
          ai += tim[l1][f2] * C8[j] - tre[l1][f2] * S8[j];
        }
        const size_t o = ((size_t)f * BATCH + b) * KDIM + k;
        xf_re[o] = ar;
        xf_im[o] = ai;
      }
    }
  }
}

// ---------------------------------------------------------------------------
// Kernel 2: expand filter via product-table gather + forward 2D DFT.
// ROW-PAIRED layout: element (k, n) at ((f*K/2 + k/2)*N + n)*2 + (k&1)
// so a GEMM B fragment is one b64 load. Canonical frequencies only.
// ---------------------------------------------------------------------------
__global__ __launch_bounds__(256) void k_expand_filter(const float* __restrict__ kern,
                                                       const int* __restrict__ mapping,
                                                       float* __restrict__ kf_re,
                                                       float* __restrict__ kf_im) {
  const int t  = blockIdx.x * 256 + threadIdx.x;  // 0 .. 65535
  const int s2 = t & 7;
  const int s1 = (t >> 3) & 7;
  const int ci = (t >> 6) & 31;
  const int co = t >> 11;
  const float* kp = kern + ((size_t)co * CIN + ci) * GDIM;
  const int*   mp = mapping + (size_t)((s1 * 8 + s2) * 64);
  const int kcol = ci * 8 + s2;
  const int ncol = co * 8 + s1;

  float tre[8][8], tim[8][8];
#pragma unroll
  for (int l1 = 0; l1 < 8; ++l1) {
    float row[8];
#pragma unroll
    for (int l2 = 0; l2 < 8; ++l2) row[l2] = kp[mp[l1 * 8 + l2]];
#pragma unroll
    for (int f2 = 0; f2 < 8; ++f2) {
      float ar = 0.f, ai = 0.f;
#pragma unroll
      for (int l2 = 0; l2 < 8; ++l2) {
        const int j = (l2 * f2) & 7;
        ar += row[l2] * C8[j];
        ai -= row[l2] * S8[j];
      }
      tre[l1][f2] = ar; tim[l1][f2] = ai;
    }
  }
#pragma unroll
  for (int f1 = 0; f1 < 8; ++f1) {
#pragma unroll
    for (int f2 = 0; f2 < 8; ++f2) {
      const int f = f1 * 8 + f2;
      if (f <= conj_freq(f)) {
        float ar = 0.f, ai = 0.f;
#pragma unroll
        for (int l1 = 0; l1 < 8; ++l1) {
          const int j = (l1 * f1) & 7;
          ar += tre[l1][f2] * C8[j] + tim[l1][f2] * S8[j];
          ai += tim[l1][f2] * C8[j] - tre[l1][f2] * S8[j];
        }
        const size_t o = ((size_t)f * (KDIM / 2) + (kcol >> 1)) * (NDIM * 2) +
                         (size_t)ncol * 2 + (kcol & 1);
        kf_re[o] = ar;
        kf_im[o] = ai;
      }
    }
  }
}

// ---------------------------------------------------------------------------
// Kernel 3: batched complex GEMM on the WMMA f32 pipe, canonical freqs only.
// Per canonical f: Y[1024,256] = X[1024,256] * K[256,256] (complex).
// Block = 256 threads (8 waves). Block tile: M=32 x N=256.
// Each wave: 2 M-tiles x 2 N-tiles -> 4 global b64 + 4 ds b64 feed
// 16 V_WMMA_F32_16X16X4_F32 per k-step.
// A tile staged in LDS via GLOBAL_LOAD_ASYNC_TO_LDS_B128 (padded stride 260).
// ---------------------------------------------------------------------------
__global__ __launch_bounds__(256) void k_cgemm_wmma(const float* __restrict__ xf_re,
                                                    const float* __restrict__ xf_im,
                                                    const float* __restrict__ kf_re,
                                                    const float* __restrict__ kf_im,
                                                    float* __restrict__ yf_re,
                                                    float* __restrict__ yf_im) {
  __shared__ float lA[2][32 * 260];

  const int tid  = threadIdx.x;
  const int lane = tid & 31;
  const int wid  = tid >> 5;
  const int f    = FCANON[blockIdx.x];       // canonical frequency
  const int b0   = blockIdx.y * 32;          // batch tile (32 rows)
  const int n0   = wid * 32;                 // wave's 32-wide N span

  // ---- stage the A tile (re + im), 32 rows x 256 K, 16B chunks ----
  {
    const float* are = xf_re + ((size_t)f * BATCH + b0) * KDIM;
    const float* aim = xf_im + ((size_t)f * BATCH + b0) * KDIM;
#pragma unroll
    for (int it = 0; it < 8; ++it) {
      const int flat = it * 256 + tid;       // 0 .. 2047
      const int m = flat >> 6;               // row 0..31
      const int c = flat & 63;               // 16B chunk 0..63
#if USE_ASYNC_LDS
      GL_ASYNC_B128(are + (size_t)m * KDIM + c * 4, &lA[0][m * 260 + c * 4]);
      GL_ASYNC_B128(aim + (size_t)m * KDIM + c * 4, &lA[1][m * 260 + c * 4]);
#else
      *(float4*)&lA[0][m * 260 + c * 4] = *(const float4*)(are + (size_t)m * KDIM + c * 4);
      *(float4*)&lA[1][m * 260 + c * 4] = *(const float4*)(aim + (size_t)m * KDIM + c * 4);
#endif
    }
  }
#if USE_ASYNC_LDS
  __builtin_amdgcn_s_wait_asynccnt(0);
#endif
  __syncthreads();

  const int m16  = lane & 15;
  const int koff = (lane >> 4) * 2;          // even -> kr even
  const int ncol = n0 + m16;
  const float* BreP0 = kf_re + (size_t)f * (KDIM / 2) * (NDIM * 2) + (size_t)ncol * 2;
  const float* BimP0 = kf_im + (size_t)f * (KDIM / 2) * (NDIM * 2) + (size_t)ncol * 2;
  const float* BreP1 = BreP0 + 32;           // +16 columns
  const float* BimP1 = BimP0 + 32;

  v8f cre00 = {}, cim00 = {}, cre10 = {}, cim10 = {};  // M-tile x, N-tile 0
  v8f cre01 = {}, cim01 = {}, cre11 = {}, cim11 = {};  // M-tile x, N-tile 1

  for (int k = 0; k < KDIM; k += 4) {
    const int kr = k + koff;                 // even
    const size_t brow = (size_t)(kr >> 1) * (NDIM * 2);
    const v2f a_re0 = *(const v2f*)&lA[0][m16 * 260 + kr];
    const v2f a_im0 = *(const v2f*)&lA[1][m16 * 260 + kr];
    const v2f a_re1 = *(const v2f*)&lA[0][(m16 + 16) * 260 + kr];
    const v2f a_im1 = *(const v2f*)&lA[1][(m16 + 16) * 260 + kr];

    const v2f b_re0 = *(const v2f*)(BreP0 + brow);
    const v2f b_im0 = *(const v2f*)(BimP0 + brow);
    const v2f b_re1 = *(const v2f*)(BreP1 + brow);
    const v2f b_im1 = *(const v2f*)(BimP1 + brow);

    if (k + 64 < KDIM) {  // prefetch ahead, alternating N-tile (folds on k&4)
      const size_t prow = (size_t)((kr + 64) >> 1) * (NDIM * 2);
      if (k & 4) {
        __builtin_prefetch(BreP1 + prow, 0, 3);
        __builtin_prefetch(BimP1 + prow, 0, 3);
      } else {
        __builtin_prefetch(BreP0 + prow, 0, 3);
        __builtin_prefetch(BimP0 + prow, 0, 3);
      }
    }

    const v2f a_imn0 = -a_im0;  // f32 WMMA has no A/B NEG modifier (ISA 7.12)
    const v2f a_imn1 = -a_im1;

    cre00 = __builtin_amdgcn_wmma_f32_16x16x4_f32(false, a_re0,  false, b_re0, (short)0, cre00, false, false);
    cre00 = __builtin_amdgcn_wmma_f32_16x16x4_f32(false, a_imn0, false, b_im0, (short)0, cre00, false, false);
    cim00 = __builtin_amdgcn_wmma_f32_16x16x4_f32(false, a_re0,  false, b_im0, (short)0, cim00, false, false);
    cim00 = __builtin_amdgcn_wmma_f32_16x16x4_f32(false, a_im0,  false, b_re0, (short)0, cim00, false, false);

    cre10 = __builtin_amdgcn_wmma_f32_16x16x4_f32(false, a_re1,  false, b_re0, (short)0, cre10, false, false);
    cre10 = __builtin_amdgcn_wmma_f32_16x16x4_f32(false, a_imn1, false, b_im0, (short)0, cre10, false, false);
    cim10 = __builtin_amdgcn_wmma_f32_16x16x4_f32(false, a_re1,  false, b_im0, (short)0, cim10, false, false);
    cim10 = __builtin_amdgcn_wmma_f32_16x16x4_f32(false, a_im1,  false, b_re0, (short)0, cim10, false, false);

    cre01 = __builtin_amdgcn_wmma_f32_16x16x4_f32(false, a_re0,  false, b_re1, (short)0, cre01, false, false);
    cre01 = __builtin_amdgcn_wmma_f32_16x16x4_f32(false, a_imn0, false, b_im1, (short)0, cre01, false, false);
    cim01 = __builtin_amdgcn_wmma_f32_16x16x4_f32(false, a_re0,  false, b_im1, (short)0, cim01, false, false);
    cim01 = __builtin_amdgcn_wmma_f32_16x16x4_f32(false, a_im0,  false, b_re1, (short)0, cim01, false, false);

    cre11 = __builtin_amdgcn_wmma_f32_16x16x4_f32(false, a_re1,  false, b_re1, (short)0, cre11, false, false);
    cre11 = __builtin_amdgcn_wmma_f32_16x16x4_f32(false, a_imn1, false, b_im1, (short)0, cre11, false, false);
    cim11 = __builtin_amdgcn_wmma_f32_16x16x4_f32(false, a_re1,  false, b_im1, (short)0, cim11, false, false);
    cim11 = __builtin_amdgcn_wmma_f32_16x16x4_f32(false, a_im1,  false, b_re1, (short)0, cim11, false, false);
  }

  const int mhi = (lane >> 4) * 8;
  float* Ore0 = yf_re + ((size_t)f * BATCH + b0 + mhi) * NDIM + ncol;
  float* Oim0 = yf_im + ((size_t)f * BATCH + b0 + mhi) * NDIM + ncol;
  float* Ore1 = Ore0 + (size_t)16 * NDIM;    // second M-tile
  float* Oim1 = Oim0 + (size_t)16 * NDIM;
#pragma unroll
  for (int r = 0; r < 8; ++r) {
    Ore0[(size_t)r * NDIM]      = cre00[r];
    Oim0[(size_t)r * NDIM]      = cim00[r];
    Ore1[(size_t)r * NDIM]      = cre10[r];
    Oim1[(size_t)r * NDIM]      = cim10[r];
    Ore0[(size_t)r * NDIM + 16] = cre01[r];
    Oim0[(size_t)r * NDIM + 16] = cim01[r];
    Ore1[(size_t)r * NDIM + 16] = cre11[r];
    Oim1[(size_t)r * NDIM + 16] = cim11[r];
  }
}

// ---------------------------------------------------------------------------
// Kernel 4: inverse 2D DFT over frequencies (+ 1/64), real part, bias add.
// Non-canonical frequencies reconstructed as conj(Y[conj(f)]) (compile-time).
// ---------------------------------------------------------------------------
__global__ __launch_bounds__(256) void k_ifft_bias(const float* __restrict__ yf_re,
                                                   const float* __restrict__ yf_im,
                                                   const float* __restrict__ bias,
                                                   float* __restrict__ out) {
  const int t  = blockIdx.x * 256 + threadIdx.x;  // 0 .. B*256-1
  const int b  = t >> 8;
  const int n  = t & 255;       // co*8 + s1
  const int co = n >> 3;
  const int s1 = n & 7;

  float tre[8][8], tim[8][8];
#pragma unroll
  for (int f1 = 0; f1 < 8; ++f1) {
    float yr[8], yi[8];
#pragma unroll
    for (int f2 = 0; f2 < 8; ++f2) {
      const int f  = f1 * 8 + f2;
      const int fc = conj_freq(f);             // folds per unrolled iteration
      const int fs = (f <= fc) ? f : fc;       // canonical source slot
      const size_t o = ((size_t)fs * BATCH + b) * NDIM + n;
      yr[f2] = yf_re[o];
      yi[f2] = (f <= fc) ? yf_im[o] : -yf_im[o];
    }
#pragma unroll
    for (int l2 = 0; l2 < 8; ++l2) {
      float ar = 0.f, ai = 0.f;
#pragma unroll
      for (int f2 = 0; f2 < 8; ++f2) {
        const int j = (f2 * l2) & 7;
        ar += yr[f2] * C8[j] - yi[f2] * S8[j];
        ai += yr[f2] * S8[j] + yi[f2] * C8[j];
      }
      tre[f1][l2] = ar;
      tim[f1][l2] = ai;
    }
  }
  const float bv = bias[co];
  float* op = out + ((size_t)b * COUT + co) * GDIM + s1;
#pragma unroll
  for (int l1 = 0; l1 < 8; ++l1) {
#pragma unroll
    for (int l2 = 0; l2 < 8; ++l2) {
      float ar = 0.f;
#pragma unroll
      for (int f1 = 0; f1 < 8; ++f1) {
        const int j = (f1 * l1) & 7;
        ar += tre[f1][l2] * C8[j] - tim[f1][l2] * S8[j];
      }
      op[(size_t)(l1 * 8 + l2) * 8] = ar * (1.f / 64.f) + bv;
    }
  }
}

// ---------------------------------------------------------------------------
extern "C" void kernel_launch(void* const* d_in, const int* in_sizes, int n_in,
                              void* d_out, int out_size, void* d_ws, size_t ws_size,
                              hipStream_t stream) {
  const float* x       = (const float*)d_in[0];  // [1024, 32, 512]
  const float* kernel  = (const float*)d_in[1];  // [32, 32, 512]
  const float* bias    = (const float*)d_in[2];  // [32, 1]
  const int*   mapping = (const int*)d_in[3];    // [8, 8, 8, 8]
  float* out = (float*)d_out;                    // [1024, 32, 512]

  float* ws = (float*)d_ws;
  const size_t XF = (size_t)FREQ * BATCH * KDIM;  // 16,777,216 floats
  const size_t KF = (size_t)FREQ * KDIM * NDIM;   //  4,194,304 floats
  float* xf_re = ws;
  float* xf_im = xf_re + XF;
  float* kf_re = xf_im + XF;
  float* kf_im = kf_re + KF;
  float* yf_re = kf_im + KF;
  float* yf_im = yf_re + XF;

  k_fft_x<<<dim3((BATCH * KDIM) / 256), dim3(256), 0, stream>>>(x, xf_re, xf_im);
  k_expand_filter<<<dim3((COUT * CIN * NS * NS) / 256), dim3(256), 0, stream>>>(
      kernel, mapping, kf_re, kf_im);
  // grid: x = canonical frequency (34), y = batch tiles (1024/32)
  k_cgemm_wmma<<<dim3(NCAN, BATCH / 32, 1), dim3(256), 0, stream>>>(
      xf_re, xf_im, kf_re, kf_im, yf_re, yf_im);
  k_ifft_bias<<<dim3((BATCH * NDIM) / 256), dim3(256), 0, stream>>>(
      yf_re, yf_im, bias, out);
}